// GCN_PYG_21698174779869
// MI455X (gfx1250) — compile-verified
//
#include <hip/hip_runtime.h>
#include <hip/hip_bf16.h>
#include <math.h>

typedef float v2f __attribute__((ext_vector_type(2)));
typedef float v8f __attribute__((ext_vector_type(8)));

#define NN 50000
#define NE 1600000
#define NF 256
#define NH 128
#define NC 64

__device__ __forceinline__ void atomic_add_f32(float* p, float v) {
    unsafeAtomicAdd(p, v);   // lowers to global_atomic_add_f32 on gfx1250
}

// ---------------- degree / norm kernels ----------------

__global__ void k_deg_init(float* __restrict__ deg, int n) {
    int i = blockIdx.x * blockDim.x + threadIdx.x;
    if (i < n) deg[i] = 1.0f;                 // self-loop weight
}

__global__ void k_deg_accum(const int* __restrict__ col, const float* __restrict__ ew,
                            float* __restrict__ deg, int e) {
    int i = blockIdx.x * blockDim.x + threadIdx.x;
    if (i < e) atomic_add_f32(&deg[col[i]], ew[i]);
}

__global__ void k_rsqrt(float* __restrict__ deg, int n) {
    int i = blockIdx.x * blockDim.x + threadIdx.x;
    if (i < n) {
        float d = deg[i];
        deg[i] = (d > 0.0f) ? rsqrtf(d) : 0.0f;
    }
}

__global__ void k_norm(const int* __restrict__ row, const int* __restrict__ col,
                       const float* __restrict__ ew, const float* __restrict__ dinv,
                       float* __restrict__ nrm, int e) {
    int i = blockIdx.x * blockDim.x + threadIdx.x;
    if (i < e) nrm[i] = dinv[row[i]] * ew[i] * dinv[col[i]];
}

// ---------------- WMMA f32 GEMM: C[M,N] = A[M,K] @ B[K,N] ----------------
// One wave32 per 16x64 output strip: 4 accumulators share one A fragment per
// K-step (4 x V_WMMA_F32_16X16X4_F32 per iteration). M % 16 == 0 (50000 =
// 3125*16), N % 64 == 0, K % 4 == 0. All exits are wave-uniform (EXEC all-1s).

__global__ __launch_bounds__(256)
void wmma_gemm_f32(const float* __restrict__ A, const float* __restrict__ B,
                   float* __restrict__ C, int M, int N, int K,
                   int ngroups, int totalTasks) {
    const int lane = threadIdx.x & 31;
    const int wave = threadIdx.x >> 5;
    const int wavesPerBlock = blockDim.x >> 5;

    int task = blockIdx.x * wavesPerBlock + wave;
    if (task >= totalTasks) return;           // wave-uniform

    const int mt = task / ngroups;            // 16-row tile index
    const int ng = task % ngroups;            // 64-col group index

    const int half = lane >> 4;               // K sub-offset selector
    const int lr   = lane & 15;               // row (A) / col (B,C)

    const float* Arow  = A + (size_t)(mt * 16 + lr) * K;
    const float* Bbase = B + (size_t)(ng * 64 + lr);

    v8f acc0 = {}, acc1 = {}, acc2 = {}, acc3 = {};
    for (int k = 0; k < K; k += 4) {
        // A frag (ISA 16x4 layout): lane lr = row, VGPR r + 2*half = K offset
        v2f a = *(const v2f*)(Arow + k + 2 * half);
        // B frags: VGPR r, lane-half h -> K = k + r + 2h, col = ng*64 + 16*j + lr
        const float* Br0 = Bbase + (size_t)(k + 2 * half) * N;
        const float* Br1 = Br0 + N;
        v2f b0 = { Br0[0],  Br1[0]  };
        v2f b1 = { Br0[16], Br1[16] };
        v2f b2 = { Br0[32], Br1[32] };
        v2f b3 = { Br0[48], Br1[48] };
        acc0 = __builtin_amdgcn_wmma_f32_16x16x4_f32(false, a, false, b0, (short)0, acc0, false, false);
        acc1 = __builtin_amdgcn_wmma_f32_16x16x4_f32(false, a, false, b1, (short)0, acc1, false, false);
        acc2 = __builtin_amdgcn_wmma_f32_16x16x4_f32(false, a, false, b2, (short)0, acc2, false, false);
        acc3 = __builtin_amdgcn_wmma_f32_16x16x4_f32(false, a, false, b3, (short)0, acc3, false, false);
    }

    // C/D layout: VGPR r, half h -> M = r + 8h, N = lr (+16 per tile)
    float* Crow = C + (size_t)(mt * 16 + 8 * half) * N + ng * 64 + lr;
#pragma unroll
    for (int r = 0; r < 8; ++r) {
        size_t o = (size_t)r * N;
        Crow[o]      = acc0[r];
        Crow[o + 16] = acc1[r];
        Crow[o + 32] = acc2[r];
        Crow[o + 48] = acc3[r];
    }
}

// ---------------- aggregation ----------------

// agg[i,:] = dinv[i]^2 * h[i,:]   (self-loop term; fully overwrites the buffer)
template <int F>
__global__ void k_self_init(const float* __restrict__ h, const float* __restrict__ dinv,
                            float* __restrict__ agg, int n) {
    int idx = blockIdx.x * blockDim.x + threadIdx.x;
    if (idx < n * F) {
        int i = idx / F;
        float di = dinv[i];
        agg[idx] = di * di * h[idx];
    }
}

// One wave32 per edge; each lane atomically adds F/32 contiguous features.
template <int F>
__global__ void k_scatter(const int* __restrict__ row, const int* __restrict__ col,
                          const float* __restrict__ nrm, const float* __restrict__ h,
                          float* __restrict__ agg, int e) {
    int tid  = blockIdx.x * blockDim.x + threadIdx.x;
    int edge = tid >> 5;
    int lane = tid & 31;
    if (edge >= e) return;
    const int V = F / 32;
    int r = row[edge];
    int c = col[edge];
    float w = nrm[edge];
    const float* hs = h   + (size_t)r * F + lane * V;
    float*       ag = agg + (size_t)c * F + lane * V;
#pragma unroll
    for (int j = 0; j < V; ++j)
        atomic_add_f32(&ag[j], w * hs[j]);
}

template <int F>
__global__ void k_sigmoid_bias(float* __restrict__ a, const float* __restrict__ b, int n) {
    int idx = blockIdx.x * blockDim.x + threadIdx.x;
    if (idx < n * F) {
        float v = a[idx] + b[idx & (F - 1)];   // F is a power of two
        a[idx] = 1.0f / (1.0f + expf(-v));
    }
}

template <int F>
__global__ void k_tanh_bias(float* __restrict__ a, const float* __restrict__ b, int n) {
    int idx = blockIdx.x * blockDim.x + threadIdx.x;
    if (idx < n * F) {
        a[idx] = tanhf(a[idx] + b[idx & (F - 1)]);
    }
}

// ---------------- launcher ----------------

extern "C" void kernel_launch(void* const* d_in, const int* in_sizes, int n_in,
                              void* d_out, int out_size, void* d_ws, size_t ws_size,
                              hipStream_t stream) {
    const float* x   = (const float*)d_in[0];   // [NN, NF]
    const int*   ei  = (const int*)d_in[1];     // [2, NE]
    const float* ew  = (const float*)d_in[2];   // [NE]
    const float* W1  = (const float*)d_in[3];   // [NF, NH]
    const float* b1  = (const float*)d_in[4];   // [NH]
    const float* W2  = (const float*)d_in[5];   // [NH, NC]
    const float* b2  = (const float*)d_in[6];   // [NC]
    float* out = (float*)d_out;                 // [NN, NC]

    const int E = in_sizes[2];                  // 1,600,000
    const int* row = ei;
    const int* col = ei + E;

    // workspace carve-up (floats): ~58 MB total
    float* ws   = (float*)d_ws;
    float* dinv = ws;                                    // NN (padded to 50048)
    float* nrm  = ws + 50048;                            // NE
    float* h1   = nrm + NE;                              // NN*NH (reused for h2)
    float* agg1 = h1 + (size_t)NN * NH;                  // NN*NH

    const int T = 256;
    dim3 blk(T);

    // 1) normalization coefficients
    k_deg_init<<<(NN + T - 1) / T, blk, 0, stream>>>(dinv, NN);
    k_deg_accum<<<(E + T - 1) / T, blk, 0, stream>>>(col, ew, dinv, E);
    k_rsqrt<<<(NN + T - 1) / T, blk, 0, stream>>>(dinv, NN);
    k_norm<<<(E + T - 1) / T, blk, 0, stream>>>(row, col, ew, dinv, nrm, E);

    // 2) layer 1: h1 = x @ W1  (50000x256 @ 256x128)
    {
        int ngroups = NH / 64;                  // 2
        int total   = (NN / 16) * ngroups;      // 6250 wave-tasks
        int blocks  = (total + 7) / 8;          // 8 waves/block
        wmma_gemm_f32<<<blocks, blk, 0, stream>>>(x, W1, h1, NN, NH, NF, ngroups, total);
    }
    k_self_init<NH><<<((size_t)NN * NH + T - 1) / T, blk, 0, stream>>>(h1, dinv, agg1, NN);
    k_scatter<NH><<<((size_t)E * 32 + T - 1) / T, blk, 0, stream>>>(row, col, nrm, h1, agg1, E);
    k_sigmoid_bias<NH><<<((size_t)NN * NH + T - 1) / T, blk, 0, stream>>>(agg1, b1, NN);

    // 3) layer 2: h2 = h @ W2  (50000x128 @ 128x64), h2 reuses h1 buffer
    float* h2 = h1;
    {
        int ngroups = NC / 64;                  // 1
        int total   = (NN / 16) * ngroups;      // 3125 wave-tasks
        int blocks  = (total + 7) / 8;
        wmma_gemm_f32<<<blocks, blk, 0, stream>>>(agg1, W2, h2, NN, NC, NH, ngroups, total);
    }
    k_self_init<NC><<<((size_t)NN * NC + T - 1) / T, blk, 0, stream>>>(h2, dinv, out, NN);
    k_scatter<NC><<<((size_t)E * 32 + T - 1) / T, blk, 0, stream>>>(row, col, nrm, h2, out, E);
    k_tanh_bias<NC><<<((size_t)NN * NC + T - 1) / T, blk, 0, stream>>>(out, b2, NN);
}